// MaskedInputLinearKernel_67645734912034
// MI455X (gfx1250) — compile-verified
//
#include <hip/hip_runtime.h>
#include <cstdint>

// ---------------------------------------------------------------------------
// Masked-input GEMV (Z=14336, N=4096, f32), single pass over W_t.
// Roofline: W_t = 235 MB read once @ 23.3 TB/s => ~10.1 us floor; FLOPs are
// negligible (~0.18 GFLOP), so the kernel is pure streaming. Each block owns
// ROWS rows; per row: gfx1250 async Global->LDS copy (ASYNCcnt, non-temporal,
// prefetch distance 2 / triple buffer for MLP), block-reduce the row norm,
// then masked FMA into per-block y partials; finish with fp32 atomics.
// ---------------------------------------------------------------------------

constexpr int ZDIM = 14336;
constexpr int NDIM = 4096;
constexpr int ROWS = 16;             // rows per workgroup
constexpr int NBLK = ZDIM / ROWS;    // 896 workgroups
constexpr int TPB  = 256;            // 8 wave32 per workgroup
constexpr int ROWBYTES = NDIM * 4;   // 16 KB per row
constexpr int NBUF = 3;              // prefetch distance 2

// 16B/lane async copy; the 24-bit inst offset applies to BOTH the LDS and the
// global address, so one per-lane offset VGPR serves all four chunks.
// Non-temporal: W_t is single-use and larger than the 192 MB L2.
#define ASYNC_B128(ldsaddr, voff, sbase, OFF)                                   \
  asm volatile("global_load_async_to_lds_b128 %0, %1, %2 offset:" OFF           \
               " th:TH_LOAD_NT"                                                 \
               :: "v"(ldsaddr), "v"(voff), "s"(sbase) : "memory")

#define PREFETCH_ROW(ldsaddr, voff, rowp)                                       \
  do {                                                                          \
    ASYNC_B128(ldsaddr, voff, rowp, "0");                                       \
    ASYNC_B128(ldsaddr, voff, rowp, "4096");                                    \
    ASYNC_B128(ldsaddr, voff, rowp, "8192");                                    \
    ASYNC_B128(ldsaddr, voff, rowp, "12288");                                   \
  } while (0)

__global__ void MaskedGemv_zero_kernel(float* __restrict__ out) {
  out[blockIdx.x * blockDim.x + threadIdx.x] = 0.0f;
}

__global__ __launch_bounds__(TPB) void MaskedGemv_fused_kernel(
    const float* __restrict__ x,      // [Z]
    const float* __restrict__ Wt,     // [Z, N] row-major
    const float* __restrict__ thr,    // [1]
    float* __restrict__ out) {        // [N], pre-zeroed
  __shared__ float buf[NBUF][NDIM];   // 3 x 16KB ring buffer
  __shared__ float red[2][TPB / 32];  // ping-pong wave partials

  const int t       = threadIdx.x;
  const int lane    = t & 31;
  const int wid     = t >> 5;
  const int rowBase = blockIdx.x * ROWS;
  const float th    = thr[0];

  const uint32_t voff = (uint32_t)t * 16u;
  // Low 32 bits of a generic LDS pointer == LDS byte offset on CDNA5.
  const uint32_t ldsBase = (uint32_t)(uintptr_t)(&buf[0][0]);
  const char*    bufBase = (const char*)(&buf[0][0]);

  // Prime the pipeline: rows 0 and 1 in flight (2 x 4 async groups).
  {
    const float* r0 = Wt + (size_t)rowBase * NDIM;
    const float* r1 = r0 + NDIM;
    PREFETCH_ROW(ldsBase + voff, voff, r0);
    PREFETCH_ROW(ldsBase + (uint32_t)ROWBYTES + voff, voff, r1);
  }

  // y partials: thread t owns columns 4t..4t+3 (+1024*j), j = 0..3
  float4 a0 = make_float4(0.f, 0.f, 0.f, 0.f);
  float4 a1 = make_float4(0.f, 0.f, 0.f, 0.f);
  float4 a2 = make_float4(0.f, 0.f, 0.f, 0.f);
  float4 a3 = make_float4(0.f, 0.f, 0.f, 0.f);

  for (int r = 0; r < ROWS; ++r) {
    // Issue prefetch for row r+2 into the ring slot last read at iter r-1
    // (its readers retired at the previous __syncthreads, which fences DScnt),
    // then retire row r's async group. ASYNCcnt decrements strictly in order:
    // <=8 leaves groups r+1, r+2 in flight.
    if (r + 2 < ROWS) {
      const float* rowp = Wt + (size_t)(rowBase + r + 2) * NDIM;
      const uint32_t l = ldsBase + (uint32_t)(((r + 2) % NBUF) * ROWBYTES) + voff;
      PREFETCH_ROW(l, voff, rowp);
      asm volatile("s_wait_asynccnt 0x8" ::: "memory");
    } else if (r + 1 < ROWS) {
      asm volatile("s_wait_asynccnt 0x4" ::: "memory");
    } else {
      asm volatile("s_wait_asynccnt 0x0" ::: "memory");
    }
    __syncthreads();  // cross-wave visibility of row r's async LDS writes

    const float4* bp = (const float4*)(bufBase + (r % NBUF) * ROWBYTES) + t;
    const float4 v0 = bp[0];
    const float4 v1 = bp[256];
    const float4 v2 = bp[512];
    const float4 v3 = bp[768];

    // Row sum-of-squares: per-thread partial over its 16 columns...
    float ss = v0.x * v0.x + v0.y * v0.y + v0.z * v0.z + v0.w * v0.w +
               v1.x * v1.x + v1.y * v1.y + v1.z * v1.z + v1.w * v1.w +
               v2.x * v2.x + v2.y * v2.y + v2.z * v2.z + v2.w * v2.w +
               v3.x * v3.x + v3.y * v3.y + v3.z * v3.z + v3.w * v3.w;
    // ...wave32 butterfly...
#pragma unroll
    for (int o = 16; o > 0; o >>= 1) ss += __shfl_xor(ss, o, 32);
    if (lane == 0) red[r & 1][wid] = ss;
    __syncthreads();  // red visible; also retires all LDS reads of slot r%NBUF
    const float* rr = red[r & 1];
    const float nrm2 = rr[0] + rr[1] + rr[2] + rr[3] +
                       rr[4] + rr[5] + rr[6] + rr[7];

    const float xz = x[rowBase + r];  // block-uniform scalar load
    if (fabsf(xz) * sqrtf(nrm2) >= th) {
      a0.x += xz * v0.x; a0.y += xz * v0.y; a0.z += xz * v0.z; a0.w += xz * v0.w;
      a1.x += xz * v1.x; a1.y += xz * v1.y; a1.z += xz * v1.z; a1.w += xz * v1.w;
      a2.x += xz * v2.x; a2.y += xz * v2.y; a2.z += xz * v2.z; a2.w += xz * v2.w;
      a3.x += xz * v3.x; a3.y += xz * v3.y; a3.z += xz * v3.z; a3.w += xz * v3.w;
    }
  }

  // Per-block partial -> global y via relaxed device-scope fp32 atomics
  // (896 * 16KB = 14.7MB of atomic traffic, ~6% of the streamed bytes).
#define ATADD(p, v) \
  __hip_atomic_fetch_add((p), (v), __ATOMIC_RELAXED, __HIP_MEMORY_SCOPE_AGENT)
  float* p0 = out + 4 * t;
  ATADD(&p0[0],        a0.x); ATADD(&p0[1],        a0.y);
  ATADD(&p0[2],        a0.z); ATADD(&p0[3],        a0.w);
  ATADD(&p0[1024],     a1.x); ATADD(&p0[1024 + 1], a1.y);
  ATADD(&p0[1024 + 2], a1.z); ATADD(&p0[1024 + 3], a1.w);
  ATADD(&p0[2048],     a2.x); ATADD(&p0[2048 + 1], a2.y);
  ATADD(&p0[2048 + 2], a2.z); ATADD(&p0[2048 + 3], a2.w);
  ATADD(&p0[3072],     a3.x); ATADD(&p0[3072 + 1], a3.y);
  ATADD(&p0[3072 + 2], a3.z); ATADD(&p0[3072 + 3], a3.w);
#undef ATADD
}

extern "C" void kernel_launch(void* const* d_in, const int* in_sizes, int n_in,
                              void* d_out, int out_size, void* d_ws, size_t ws_size,
                              hipStream_t stream) {
  (void)in_sizes; (void)n_in; (void)d_ws; (void)ws_size; (void)out_size;
  const float* x   = (const float*)d_in[0];
  const float* Wt  = (const float*)d_in[1];
  const float* thr = (const float*)d_in[2];
  float* out = (float*)d_out;

  hipLaunchKernelGGL(MaskedGemv_zero_kernel, dim3(NDIM / TPB), dim3(TPB), 0,
                     stream, out);
  hipLaunchKernelGGL(MaskedGemv_fused_kernel, dim3(NBLK), dim3(TPB), 0, stream,
                     x, Wt, thr, out);
}